// Net_29283087024926
// MI455X (gfx1250) — compile-verified
//
#include <hip/hip_runtime.h>
#include <hip/hip_bf16.h>

// GCN pipeline for MI455X (gfx1250, wave32).
// GEMMs via V_WMMA_F32_16X16X4_F32, 64x64 macro-tile per wave:
//   16 independent accumulators -> XDL pipe stays busy, 4x A/B fragment reuse
//   -> ~4x less L2 traffic than a 16x16-per-wave GEMM.
// Edge aggregation via L2-resident f32 atomics (all feature mats << 192MB L2).
// Linear-algebra reordering: layer3 GEMM-before-scatter (scatter 64 not 128 wide),
// layer5 scatter-before-GEMM (scatter 64 not 1024 wide).

#define GN      50000
#define GE      600000
#define GG      100
#define F_IN    50
#define F_INP   52     // padded to multiple of 4 for WMMA K-loop
#define SLOPE   0.01f

typedef float v2f __attribute__((ext_vector_type(2)));
typedef float v8f __attribute__((ext_vector_type(8)));

static __host__ inline int cdiv(long a, int b) { return (int)((a + b - 1) / b); }

// ---------------- elementwise / utility kernels ----------------

__global__ void k_zero(float* __restrict__ p, long n) {
    long i = (long)blockIdx.x * blockDim.x + threadIdx.x;
    if (i < n) p[i] = 0.0f;
}

__global__ void k_degrees(const int* __restrict__ src, const int* __restrict__ dst,
                          float* __restrict__ deg_out, float* __restrict__ deg_in, int e_cnt) {
    int e = blockIdx.x * blockDim.x + threadIdx.x;
    if (e < e_cnt) {
        atomicAdd(&deg_out[src[e]], 1.0f);
        atomicAdd(&deg_in[dst[e]], 1.0f);
    }
}

// in: inv_mean holds deg_in, rs_out holds deg_out. out: inv_mean=1/max(di,1),
// rs_in=rsqrt(max(di,1)), rs_out=rsqrt(max(do,1))
__global__ void k_finalize_deg(float* __restrict__ inv_mean, float* __restrict__ rs_in,
                               float* __restrict__ rs_out, int n) {
    int i = blockIdx.x * blockDim.x + threadIdx.x;
    if (i < n) {
        float di = fmaxf(inv_mean[i], 1.0f);
        float dd = fmaxf(rs_out[i], 1.0f);
        inv_mean[i] = 1.0f / di;
        rs_in[i]    = rsqrtf(di);
        rs_out[i]   = rsqrtf(dd);
    }
}

__global__ void k_pad_w1(const float* __restrict__ W1, float* __restrict__ W1p, int n) {
    int i = blockIdx.x * blockDim.x + threadIdx.x;   // over F_INP*128
    if (i < n) {
        int row = i >> 7;
        W1p[i] = (row < F_IN) ? W1[i] : 0.0f;        // same row-major layout for row<50
    }
}

// scatter-add x[src[e], :dim] into agg[dst[e], :dim], float2 chunks (dim even)
__global__ void k_scatter2(const float* __restrict__ x, int ldx,
                           const int* __restrict__ src, const int* __restrict__ dst,
                           float* __restrict__ agg, int ldagg, int nch, long total) {
    long t = (long)blockIdx.x * blockDim.x + threadIdx.x;
    if (t >= total) return;
    long e = t / nch;
    int  f = (int)(t - e * nch) * 2;
    int  s = src[e], d = dst[e];
    float2 v = *(const float2*)(x + (size_t)s * ldx + f);
    float* p = agg + (size_t)d * ldagg + f;
    atomicAdd(p + 0, v.x);
    atomicAdd(p + 1, v.y);
}

// scatter-add, float4 chunks (dim multiple of 4, rows 16B aligned)
__global__ void k_scatter4(const float* __restrict__ x, int ldx,
                           const int* __restrict__ src, const int* __restrict__ dst,
                           float* __restrict__ agg, int ldagg, int nch, long total) {
    long t = (long)blockIdx.x * blockDim.x + threadIdx.x;
    if (t >= total) return;
    long e = t / nch;
    int  f = (int)(t - e * nch) * 4;
    int  s = src[e], d = dst[e];
    float4 v = *(const float4*)(x + (size_t)s * ldx + f);
    float* p = agg + (size_t)d * ldagg + f;
    atomicAdd(p + 0, v.x);
    atomicAdd(p + 1, v.y);
    atomicAdd(p + 2, v.z);
    atomicAdd(p + 3, v.w);
}

__global__ void k_scale_rows(const float* __restrict__ x, const float* __restrict__ s,
                             float* __restrict__ y, int dim, long total) {
    long t = (long)blockIdx.x * blockDim.x + threadIdx.x;
    if (t < total) y[t] = x[t] * s[t / dim];
}

// y = s[row]*x + b[col]
__global__ void k_scale_bias(const float* __restrict__ x, const float* __restrict__ s,
                             const float* __restrict__ b, float* __restrict__ y,
                             int dim, long total) {
    long t = (long)blockIdx.x * blockDim.x + threadIdx.x;
    if (t < total) {
        long row = t / dim;
        int  f   = (int)(t - row * dim);
        y[t] = x[t] * s[row] + b[f];
    }
}

__global__ void k_count(const int* __restrict__ gid, float* __restrict__ cnt, int n) {
    int i = blockIdx.x * blockDim.x + threadIdx.x;
    if (i < n) atomicAdd(&cnt[gid[i]], 1.0f);
}

__global__ void k_pool_sum(const float* __restrict__ x, const int* __restrict__ gid,
                           float* __restrict__ pooled, long total) {  // total = N*64
    long t = (long)blockIdx.x * blockDim.x + threadIdx.x;
    if (t < total) {
        long node = t >> 6;
        int  f    = (int)(t & 63);
        atomicAdd(&pooled[(size_t)gid[node] * 64 + f], x[t]);
    }
}

__global__ void k_inv_cnt(float* __restrict__ cnt, int g) {
    int i = blockIdx.x * blockDim.x + threadIdx.x;
    if (i < g) cnt[i] = 1.0f / fmaxf(cnt[i], 1.0f);
}

__global__ void k_bcast(const float* __restrict__ pooled, const float* __restrict__ icnt,
                        const int* __restrict__ gid, float* __restrict__ y, long total) {
    long t = (long)blockIdx.x * blockDim.x + threadIdx.x;
    if (t < total) {
        long node = t >> 6;
        int  f    = (int)(t & 63);
        int  g    = gid[node];
        y[t] = pooled[(size_t)g * 64 + f] * icnt[g];
    }
}

// ---------------- fp32 WMMA GEMM, 64x64 macro-tile per wave ----------------
// out[M x Nout] = act( diag(scale) * A[M x K] @ W[K x Nout] + bias )
// One wave computes a 64x64 tile = 4x4 grid of 16x16 WMMA subtiles
// (16 independent accumulators; A/B fragments each reused 4x).
// K multiple of 4; Nout multiple of 64; M multiple of 16 (tail M-subtiles of the
// last macro-row are clamped for loads and skipped for stores).
// Fragment layouts per ISA 7.12.2 (32-bit, wave32):
//   A 16x4 : all lanes hold row M=lane&15; lanes 0-15 -> K=k,k+1, lanes 16-31 -> K=k+2,k+3
//   B 4x16 : col N=lane&15;  lanes 0-15 -> K=k,k+1, lanes 16-31 -> K=k+2,k+3
//   D vgpr r: lanes 0-15 -> M=r, lanes 16-31 -> M=8+r; N=lane&15
__global__ void __launch_bounds__(32)
k_gemm_wmma_f32(const float* __restrict__ A, int lda, int K,
                const float* __restrict__ scale,
                const float* __restrict__ W,
                const float* __restrict__ bias,
                float* __restrict__ out, int Nout, int leaky, int M) {
    const int lane = threadIdx.x;
    const int lm   = lane & 15;
    const int half = lane >> 4;

    const long mBase = (long)blockIdx.x * 64;
    const int  nBase = blockIdx.y * 64;

    const float* ap[4];
    float s[4];
    bool  mOK[4];
#pragma unroll
    for (int i = 0; i < 4; ++i) {
        long rb = mBase + i * 16;
        mOK[i]  = rb < (long)M;              // uniform (M % 16 == 0)
        long row = (mOK[i] ? rb : 0) + lm;   // clamp keeps loads in-bounds
        ap[i] = A + row * (long)lda + half * 2;
        s[i]  = scale ? scale[row] : 1.0f;
    }

    const float* bp[4];
    float bb[4];
#pragma unroll
    for (int j = 0; j < 4; ++j) {
        int col = nBase + j * 16 + lm;
        bp[j] = W + (size_t)(half * 2) * Nout + col;
        bb[j] = bias ? bias[col] : 0.0f;
    }

    v8f acc[16];
#pragma unroll
    for (int t = 0; t < 16; ++t) acc[t] = (v8f){0.f, 0.f, 0.f, 0.f, 0.f, 0.f, 0.f, 0.f};

    for (int k = 0; k < K; k += 4) {
        v2f a[4], b[4];
#pragma unroll
        for (int i = 0; i < 4; ++i) {
            a[i] = *(const v2f*)(ap[i] + k);
            a[i] *= s[i];
        }
#pragma unroll
        for (int j = 0; j < 4; ++j) {
            b[j].x = bp[j][(size_t)k * Nout];
            b[j].y = bp[j][(size_t)(k + 1) * Nout];
        }
#pragma unroll
        for (int i = 0; i < 4; ++i)
#pragma unroll
            for (int j = 0; j < 4; ++j)
                acc[i * 4 + j] = __builtin_amdgcn_wmma_f32_16x16x4_f32(
                    /*neg_a=*/false, a[i], /*neg_b=*/false, b[j],
                    /*c_mod=*/(short)0, acc[i * 4 + j],
                    /*reuse_a=*/false, /*reuse_b=*/false);
    }

#pragma unroll
    for (int i = 0; i < 4; ++i) {
        if (!mOK[i]) continue;
        const long orow0 = mBase + i * 16 + half * 8;
#pragma unroll
        for (int j = 0; j < 4; ++j) {
            const int col = nBase + j * 16 + lm;
#pragma unroll
            for (int r = 0; r < 8; ++r) {
                float v = acc[i * 4 + j][r] + bb[j];
                if (leaky) v = (v > 0.0f) ? v : SLOPE * v;
                out[(orow0 + r) * (long)Nout + col] = v;
            }
        }
    }
}

// ---------------- driver ----------------

extern "C" void kernel_launch(void* const* d_in, const int* in_sizes, int n_in,
                              void* d_out, int out_size, void* d_ws, size_t ws_size,
                              hipStream_t stream) {
    const float* features = (const float*)d_in[0];
    const float* W1  = (const float*)d_in[1];
    const float* b1  = (const float*)d_in[2];
    const float* Wg  = (const float*)d_in[3];
    const float* bg  = (const float*)d_in[4];
    const float* W2  = (const float*)d_in[5];
    const float* b2  = (const float*)d_in[6];
    const float* Wg2 = (const float*)d_in[7];
    const float* bg2 = (const float*)d_in[8];
    const float* W3  = (const float*)d_in[9];
    const float* b3  = (const float*)d_in[10];
    const int*   src = (const int*)d_in[11];
    const int*   dst = (const int*)d_in[12];
    const int*   gid = (const int*)d_in[13];
    float* out = (float*)d_out;

    // workspace carve-up (floats, all chunks multiple of 4 floats -> 16B aligned)
    float* w = (float*)d_ws;
    float* inv_mean = w; w += GN;            // holds deg_in during accumulation
    float* rs_in    = w; w += GN;
    float* rs_out   = w; w += GN;            // holds deg_out during accumulation
    float* W1p      = w; w += F_INP * 128;
    float* agg50    = w; w += (size_t)GN * F_INP;
    float* B1       = w; w += (size_t)GN * 128;
    float* B2       = w; w += (size_t)GN * 128;
    float* B3       = w; w += (size_t)GN * 128;
    float* B4       = w; w += (size_t)GN * 128;
    float* pooled   = w; w += GG * 64;
    float* cnt      = w; w += GG;

    const int BS = 256;
    const long N128 = (long)GN * 128, N64 = (long)GN * 64;
    const int MG = cdiv(GN, 64);             // macro-tile rows (782)

    // --- degrees ---
    k_zero<<<cdiv(GN, BS), BS, 0, stream>>>(inv_mean, GN);
    k_zero<<<cdiv(GN, BS), BS, 0, stream>>>(rs_out, GN);
    k_degrees<<<cdiv(GE, BS), BS, 0, stream>>>(src, dst, rs_out, inv_mean, GE);
    k_finalize_deg<<<cdiv(GN, BS), BS, 0, stream>>>(inv_mean, rs_in, rs_out, GN);

    // --- layer 1: gcn_layer(features, W1) + leaky ---
    k_zero<<<cdiv((long)GN * F_INP, BS), BS, 0, stream>>>(agg50, (long)GN * F_INP);
    k_pad_w1<<<cdiv(F_INP * 128, BS), BS, 0, stream>>>(W1, W1p, F_INP * 128);
    {
        long total = (long)GE * (F_IN / 2);
        k_scatter2<<<cdiv(total, BS), BS, 0, stream>>>(features, F_IN, src, dst,
                                                       agg50, F_INP, F_IN / 2, total);
    }
    k_gemm_wmma_f32<<<dim3(MG, 128 / 64), 32, 0, stream>>>(
        agg50, F_INP, F_INP, inv_mean, W1p, b1, B1, 128, 1, GN);

    // --- layer 2: graph_conv(B1, Wg) 128->128 ---
    k_scale_rows<<<cdiv(N128, BS), BS, 0, stream>>>(B1, rs_out, B2, 128, N128);
    k_zero<<<cdiv(N128, BS), BS, 0, stream>>>(B3, N128);
    {
        long total = (long)GE * (128 / 4);
        k_scatter4<<<cdiv(total, BS), BS, 0, stream>>>(B2, 128, src, dst, B3, 128, 128 / 4, total);
    }
    k_gemm_wmma_f32<<<dim3(MG, 128 / 64), 32, 0, stream>>>(
        B3, 128, 128, rs_in, Wg, bg, B4, 128, 1, GN);

    // --- layer 3: gcn_layer(B4, W2) 128->64, GEMM-first (scatter at 64 wide) ---
    k_gemm_wmma_f32<<<dim3(MG, 64 / 64), 32, 0, stream>>>(
        B4, 128, 128, nullptr, W2, nullptr, B1, 64, 0, GN);        // t = h2 @ W2
    k_zero<<<cdiv(N64, BS), BS, 0, stream>>>(B2, N64);
    {
        long total = (long)GE * (64 / 4);
        k_scatter4<<<cdiv(total, BS), BS, 0, stream>>>(B1, 64, src, dst, B2, 64, 64 / 4, total);
    }
    k_scale_bias<<<cdiv(N64, BS), BS, 0, stream>>>(B2, inv_mean, b2, B3, 64, N64); // h3

    // --- avg-pool per graph, broadcast back ---
    k_zero<<<cdiv(GG * 64, BS), BS, 0, stream>>>(pooled, GG * 64);
    k_zero<<<1, BS, 0, stream>>>(cnt, GG);
    k_count<<<cdiv(GN, BS), BS, 0, stream>>>(gid, cnt, GN);
    k_pool_sum<<<cdiv(N64, BS), BS, 0, stream>>>(B3, gid, pooled, N64);
    k_inv_cnt<<<1, BS, 0, stream>>>(cnt, GG);
    k_bcast<<<cdiv(N64, BS), BS, 0, stream>>>(pooled, cnt, gid, B1, N64);  // x4

    // --- layer 4: graph_conv(B1, Wg2) 64->64 ---
    k_scale_rows<<<cdiv(N64, BS), BS, 0, stream>>>(B1, rs_out, B2, 64, N64);
    k_zero<<<cdiv(N64, BS), BS, 0, stream>>>(B3, N64);
    {
        long total = (long)GE * (64 / 4);
        k_scatter4<<<cdiv(total, BS), BS, 0, stream>>>(B2, 64, src, dst, B3, 64, 64 / 4, total);
    }
    k_gemm_wmma_f32<<<dim3(MG, 64 / 64), 32, 0, stream>>>(
        B3, 64, 64, rs_in, Wg2, bg2, B4, 64, 1, GN);

    // --- layer 5: gcn_layer(B4, W3) 64->1024, aggregate-first (scatter at 64 wide) ---
    k_zero<<<cdiv(N64, BS), BS, 0, stream>>>(B1, N64);
    {
        long total = (long)GE * (64 / 4);
        k_scatter4<<<cdiv(total, BS), BS, 0, stream>>>(B4, 64, src, dst, B1, 64, 64 / 4, total);
    }
    k_gemm_wmma_f32<<<dim3(MG, 1024 / 64), 32, 0, stream>>>(
        B1, 64, 64, inv_mean, W3, b3, out, 1024, 0, GN);
}